// Entailment_loss_20976620273876
// MI455X (gfx1250) — compile-verified
//
#include <hip/hip_runtime.h>

#define D_DIM 128
#define K_CONST 0.1f
#define GAMMA_C 1.0f

typedef float v2f __attribute__((ext_vector_type(2)));
typedef float v8f __attribute__((ext_vector_type(8)));

// ws layout (floats):
//   [0..127]   colsum  (sum over rows of x[:,d])
//   [128]      sum_nx2 (sum_i ||x_i||^2)
//   [129]      acc     (global loss accumulator)
//   [256 .. 256+N)                nx2[i]  = ||x_i||^2
//   [256+N .. 256+N+M)            np2[j]  = ||p_j||^2
//   [256+N+M .. 256+N+2M)         f[j]    = ||p_j|| * sqrt(sum_sq_dist_j)
//   [256+N+2M .. 256+N+3M)        psi[j]

__global__ void k_zero(float* __restrict__ ws) {
    int t = threadIdx.x;
    ws[t] = 0.0f;  // covers colsum[0..127], sum_nx2, acc, padding up to 256
}

__global__ void k_xstats(const float* __restrict__ x, float* __restrict__ ws, int N) {
    __shared__ float scol[D_DIM];
    __shared__ float sred[128];
    const int t = threadIdx.x;
    const int r = blockIdx.x * 128 + t;
    scol[t] = 0.0f;
    __syncthreads();

    const float4* xr = (const float4*)(x + (size_t)r * D_DIM);
    float nx2 = 0.0f;
    // rotate float4-chunk start per thread to spread LDS-atomic bank traffic
    for (int q = 0; q < 32; ++q) {
        int qi = (q + t) & 31;
        float4 v = xr[qi];
        nx2 += v.x * v.x + v.y * v.y + v.z * v.z + v.w * v.w;
        atomicAdd(&scol[qi * 4 + 0], v.x);
        atomicAdd(&scol[qi * 4 + 1], v.y);
        atomicAdd(&scol[qi * 4 + 2], v.z);
        atomicAdd(&scol[qi * 4 + 3], v.w);
    }
    ws[256 + r] = nx2;
    sred[t] = nx2;
    __syncthreads();

    atomicAdd(&ws[t], scol[t]);  // global colsum

    for (int s = 64; s > 0; s >>= 1) {
        if (t < s) sred[t] += sred[t + s];
        __syncthreads();
    }
    if (t == 0) atomicAdd(&ws[128], sred[0]);
}

__global__ void k_pstats(const float* __restrict__ p, float* __restrict__ ws,
                         int N, int M) {
    const int j = blockIdx.x * 128 + threadIdx.x;
    const float4* pr = (const float4*)(p + (size_t)j * D_DIM);
    const float4* cs = (const float4*)(ws);
    float np2 = 0.0f, dcs = 0.0f;
    for (int q = 0; q < 32; ++q) {
        float4 v = pr[q];
        float4 c = cs[q];
        np2 += v.x * v.x + v.y * v.y + v.z * v.z + v.w * v.w;
        dcs += v.x * c.x + v.y * c.y + v.z * c.z + v.w * c.w;
    }
    float sum_nx2 = ws[128];
    float ssd = sum_nx2 + (float)N * np2 - 2.0f * dcs;   // sum_i ||x_i - p_j||^2
    float npn = sqrtf(np2);
    const int base = 256 + N;
    ws[base + j]         = np2;
    ws[base + M + j]     = npn * sqrtf(fmaxf(ssd, 0.0f));
    ws[base + 2 * M + j] = asinf(K_CONST * (1.0f - np2) / npn);
}

__global__ void __launch_bounds__(256)
k_main(const float* __restrict__ x, const float* __restrict__ p,
       const int* __restrict__ labels, float* __restrict__ ws, int N, int M) {
    const int tid  = threadIdx.x;
    const int w    = tid >> 5;          // wave id 0..7 (wave32)
    const int lane = tid & 31;
    const int half = lane >> 4;         // K-pair / M-half selector
    const int l16  = lane & 15;

    const int i0 = blockIdx.x * 32 + (w >> 2) * 16;   // 2 wave-rows
    const int j0 = blockIdx.y * 64 + (w & 3) * 16;    // 4 wave-cols

    const float* __restrict__ nx2v = ws + 256;
    const float* __restrict__ np2v = ws + 256 + N;
    const float* __restrict__ fv   = np2v + M;
    const float* __restrict__ psiv = fv + M;

    // ISA 32-bit 16x4 A layout: lane = M (0..15), half selects K pair {0,1}/{2,3}
    // B (4x16) mirrored: lane = N, half selects K pair. Each lane loads float2.
    const float* arow = x + (size_t)(i0 + l16) * D_DIM + half * 2;
    const float* brow = p + (size_t)(j0 + l16) * D_DIM + half * 2;

    v8f c = {0.f, 0.f, 0.f, 0.f, 0.f, 0.f, 0.f, 0.f};
#pragma unroll
    for (int k = 0; k < D_DIM; k += 4) {
        v2f a = *(const v2f*)(arow + k);
        v2f b = *(const v2f*)(brow + k);
        c = __builtin_amdgcn_wmma_f32_16x16x4_f32(
                /*neg_a=*/false, a, /*neg_b=*/false, b,
                /*c_mod=*/(short)0, c, /*reuse_a=*/false, /*reuse_b=*/false);
    }

    // Column (j) is fixed per lane in the C/D layout.
    const int j   = j0 + l16;
    const float np2j = np2v[j];
    const float fj   = fv[j];
    const float psij = psiv[j];
    const float c1   = 1.0f + np2j;

    float part = 0.0f;
#pragma unroll
    for (int r = 0; r < 8; ++r) {
        const int i = i0 + r + half * 8;   // C layout: VGPR r -> M = r (+8 for upper half)
        const float dotv = c[r];
        const float nx2i = nx2v[i];
        const int   li   = labels[i];
        const float num  = dotv * c1 - np2j * (1.0f + nx2i);
        const float t2   = 1.0f + np2j * nx2i - 2.0f * dotv;
        const float den  = fj * sqrtf(fmaxf(t2, 1e-30f));
        const float ratio = fminf(1.0f, fmaxf(-1.0f, num / den));
        const float ang  = acosf(ratio);
        const float a    = fmaxf(0.0f, ang - psij);
        // per-element contribution: label column contributes a_ij (positive term
        // replaces its excluded negative term), others contribute relu(gamma - a).
        part += (j == li) ? a : fmaxf(0.0f, GAMMA_C - a);
    }

    __shared__ float sred[256];
    sred[tid] = part;
    __syncthreads();
    for (int s = 128; s > 0; s >>= 1) {
        if (tid < s) sred[tid] += sred[tid + s];
        __syncthreads();
    }
    if (tid == 0) atomicAdd(&ws[129], sred[0]);
}

__global__ void k_final(const float* __restrict__ ws, float* __restrict__ out, int N) {
    out[0] = ws[129] / (float)N;
}

extern "C" void kernel_launch(void* const* d_in, const int* in_sizes, int n_in,
                              void* d_out, int out_size, void* d_ws, size_t ws_size,
                              hipStream_t stream) {
    const float* x      = (const float*)d_in[0];
    const float* p      = (const float*)d_in[1];
    const int*   labels = (const int*)d_in[2];
    float* ws  = (float*)d_ws;
    float* out = (float*)d_out;

    const int N = in_sizes[0] / D_DIM;  // 16384
    const int M = in_sizes[1] / D_DIM;  // 2048

    k_zero<<<1, 256, 0, stream>>>(ws);
    k_xstats<<<N / 128, 128, 0, stream>>>(x, ws, N);
    k_pstats<<<M / 128, 128, 0, stream>>>(p, ws, N, M);
    dim3 grid(N / 32, M / 64);
    k_main<<<grid, 256, 0, stream>>>(x, p, labels, ws, N, M);
    k_final<<<1, 1, 0, stream>>>(ws, out, N);
}